// HyperbolicGraphMatching_90426241450714
// MI455X (gfx1250) — compile-verified
//
#include <hip/hip_runtime.h>
#include <hip/hip_bf16.h>
#include <math.h>

#define B_    8
#define NS    4096
#define NT    4096
#define CH    128
#define KTOP  32
#define ROWS  16           // source rows per workgroup
#define PITCH (NT + 4)     // LDS row pitch in floats (16B-aligned: 4100*4 % 16 == 0)
#define NCHUNK 128         // 4096 / 32 chunks per row

typedef float v2f __attribute__((ext_vector_type(2)));
typedef float v8f __attribute__((ext_vector_type(8)));

__launch_bounds__(256, 1)
__global__ void hgm_fused_kernel(const float* __restrict__ hs,
                                 const float* __restrict__ ht,
                                 float* __restrict__ out_val,
                                 float* __restrict__ out_idx) {
  extern __shared__ float smem[];
  float* keys   = smem;                                 // ROWS * PITCH
  float* cmax   = smem + ROWS * PITCH;                  // ROWS * NCHUNK
  int*   topIdx = (int*)(cmax + ROWS * NCHUNK);         // ROWS * KTOP
  float* shat   = (float*)(topIdx + ROWS * KTOP);       // ROWS * KTOP

  const int tid  = threadIdx.x;
  const int wave = tid >> 5;
  const int lane = tid & 31;
  const int b    = blockIdx.x / (NS / ROWS);
  const int m0   = (blockIdx.x % (NS / ROWS)) * ROWS;

  // ---------------- Stage 1: f32 WMMA GEMM, double-buffered B ------------
  // A frag (16x4 f32): lanes 0-15 rows M=0..15 K pair {0,1}; lanes 16-31 K {2,3}.
  const int arow  = lane & 15;
  const int khalf = (lane >> 4) * 2;
  const int ncol  = lane & 15;

  v2f afrag[32];
  {
    const float* aptr = hs + ((size_t)b * NS + (size_t)(m0 + arow)) * CH;
#pragma unroll
    for (int ks = 0; ks < 32; ++ks) {
      v2f a = *(const v2f*)(aptr + ks * 4 + khalf);
      if (ks == 0 && khalf == 0) a.x = -a.x;     // Lorentz: negate component 0
      afrag[ks] = a;
    }
  }

  const float* htb = ht + ((size_t)b * NT + (size_t)ncol) * CH + khalf;
  const int rbase = (lane >> 4) * 8;

  v2f bufA[32], bufB[32];
  {
    const float* q = htb + (size_t)(wave * 512) * CH;
#pragma unroll
    for (int ks = 0; ks < 32; ++ks) bufA[ks] = *(const v2f*)(q + ks * 4);
  }

  for (int t = 0; t < 32; t += 2) {
    const int n0 = wave * 512 + t * 16;
    // prefetch tile t+1 into bufB (overlaps WMMA chains below)
    {
      const float* q = htb + (size_t)(n0 + 16) * CH;
#pragma unroll
      for (int ks = 0; ks < 32; ++ks) bufB[ks] = *(const v2f*)(q + ks * 4);
    }
    {
      // two independent accumulator chains (dep distance 2), merged at end
      v8f acc0 = {0.f,0.f,0.f,0.f,0.f,0.f,0.f,0.f};
      v8f acc1 = {0.f,0.f,0.f,0.f,0.f,0.f,0.f,0.f};
#pragma unroll
      for (int ks = 0; ks < 16; ++ks) {
        acc0 = __builtin_amdgcn_wmma_f32_16x16x4_f32(false, afrag[ks], false, bufA[ks],
                                                     (short)0, acc0, false, false);
        acc1 = __builtin_amdgcn_wmma_f32_16x16x4_f32(false, afrag[ks + 16], false, bufA[ks + 16],
                                                     (short)0, acc1, false, false);
      }
#pragma unroll
      for (int v = 0; v < 8; ++v)
        keys[(rbase + v) * PITCH + n0 + ncol] = fminf(acc0[v] + acc1[v], -1.0f);
    }
    // prefetch tile t+2 into bufA
    if (t + 2 < 32) {
      const float* q = htb + (size_t)(n0 + 32) * CH;
#pragma unroll
      for (int ks = 0; ks < 32; ++ks) bufA[ks] = *(const v2f*)(q + ks * 4);
    }
    {
      v8f acc0 = {0.f,0.f,0.f,0.f,0.f,0.f,0.f,0.f};
      v8f acc1 = {0.f,0.f,0.f,0.f,0.f,0.f,0.f,0.f};
#pragma unroll
      for (int ks = 0; ks < 16; ++ks) {
        acc0 = __builtin_amdgcn_wmma_f32_16x16x4_f32(false, afrag[ks], false, bufB[ks],
                                                     (short)0, acc0, false, false);
        acc1 = __builtin_amdgcn_wmma_f32_16x16x4_f32(false, afrag[ks + 16], false, bufB[ks + 16],
                                                     (short)0, acc1, false, false);
      }
#pragma unroll
      for (int v = 0; v < 8; ++v)
        keys[(rbase + v) * PITCH + n0 + 16 + ncol] = fminf(acc0[v] + acc1[v], -1.0f);
    }
  }
  __syncthreads();

  // ---------------- Stage 2a: per-chunk maxima (32-wide chunks) ----------
  for (int c = tid; c < ROWS * NCHUNK; c += 256) {
    const int r = c >> 7, ch = c & (NCHUNK - 1);
    const float* p = keys + r * PITCH + ch * 32;
    float m = -3.4e38f;
#pragma unroll
    for (int i = 0; i < 8; ++i) {
      float4 v = *(const float4*)(p + i * 4);
      m = fmaxf(m, fmaxf(fmaxf(v.x, v.y), fmaxf(v.z, v.w)));
    }
    cmax[r * NCHUNK + ch] = m;
  }
  __syncthreads();

  // ---------------- Stage 2b: top-32 per row, hierarchical argmax --------
  for (int r = wave * 2; r < wave * 2 + 2; ++r) {
    float* krw = keys + r * PITCH;
    float* cm  = cmax + r * NCHUNK;
    for (int it = 0; it < KTOP; ++it) {
      // level 1: best chunk (ties -> lowest chunk index)
      float bv = -3.4e38f;
      int   bc = NCHUNK;
#pragma unroll
      for (int j = 0; j < 4; ++j) {
        const int c = lane * 4 + j;                 // ascending: first hit wins
        float v = cm[c];
        if (v > bv) { bv = v; bc = c; }
      }
#pragma unroll
      for (int m = 16; m >= 1; m >>= 1) {
        float ov = __shfl_xor(bv, m, 32);
        int   oc = __shfl_xor(bc, m, 32);
        if (ov > bv || (ov == bv && oc < bc)) { bv = ov; bc = oc; }
      }
      // level 2: best element in winning chunk (ties -> lowest index)
      const int base = bc * 32;
      float ev = krw[base + lane];
      int   ei = base + lane;
#pragma unroll
      for (int m = 16; m >= 1; m >>= 1) {
        float ov = __shfl_xor(ev, m, 32);
        int   oi = __shfl_xor(ei, m, 32);
        if (ov > ev || (ov == ev && oi < ei)) { ev = ov; ei = oi; }
      }
      if (lane == 0) topIdx[r * KTOP + it] = ei;
      krw[ei] = -3.4e38f;                           // uniform store, mask winner
      // refresh this chunk's max
      float nv = krw[base + lane];
#pragma unroll
      for (int m = 16; m >= 1; m >>= 1) nv = fmaxf(nv, __shfl_xor(nv, m, 32));
      cm[bc] = nv;                                  // uniform store
    }
  }
  __syncthreads();

  // ---------------- Stage 3: exact recompute of candidates + transform ----
  for (int p = tid; p < ROWS * KTOP; p += 256) {
    const int r   = p / KTOP;
    const int idx = topIdx[p];
    const float4* a4 = (const float4*)(hs + ((size_t)b * NS + (size_t)(m0 + r)) * CH);
    const float4* b4 = (const float4*)(ht + ((size_t)b * NT + (size_t)idx) * CH);
    float s = 0.f;
#pragma unroll
    for (int c = 0; c < CH / 4; ++c) {
      float4 av = a4[c], bv = b4[c];
      s += av.x * bv.x + av.y * bv.y + av.z * bv.z + av.w * bv.w;
    }
    s -= 2.0f * a4[0].x * b4[0].x;
    float theta = fmaxf(-s, 1.0f);
    float ar = __logf(theta + sqrtf(fmaxf(theta * theta - 1.0f, 0.0f)));
    shat[p] = -ar * ar;
  }
  __syncthreads();

  // ---------------- Stage 4: softmax over k, emit COO values + indices ----
  if (tid < ROWS) {
    const int r = tid;
    float mx = -3.4e38f;
#pragma unroll
    for (int k = 0; k < KTOP; ++k) mx = fmaxf(mx, shat[r * KTOP + k]);
    float e[KTOP];
    float sum = 0.f;
#pragma unroll
    for (int k = 0; k < KTOP; ++k) {
      e[k] = __expf(shat[r * KTOP + k] - mx);
      sum += e[k];
    }
    const float inv = 1.0f / sum;
    const size_t base = ((size_t)b * NS + (size_t)(m0 + r)) * KTOP;
#pragma unroll
    for (int k = 0; k < KTOP; ++k) {
      out_val[base + k] = e[k] * inv;
      out_idx[base + k] = (float)topIdx[r * KTOP + k];
    }
  }
}

extern "C" void kernel_launch(void* const* d_in, const int* in_sizes, int n_in,
                              void* d_out, int out_size, void* d_ws, size_t ws_size,
                              hipStream_t stream) {
  (void)in_sizes; (void)n_in; (void)out_size; (void)d_ws; (void)ws_size;
  const float* hs = (const float*)d_in[0];
  const float* ht = (const float*)d_in[1];
  float* out_val = (float*)d_out;
  float* out_idx = out_val + (size_t)B_ * NS * KTOP;

  const dim3 grid(B_ * (NS / ROWS));
  const dim3 block(256);
  const size_t smem_bytes =
      (size_t)ROWS * PITCH * sizeof(float) +        // keys slab (262,400 B)
      (size_t)ROWS * NCHUNK * sizeof(float) +       // chunk maxima (8 KB)
      (size_t)ROWS * KTOP * sizeof(int) +           // top-k indices
      (size_t)ROWS * KTOP * sizeof(float);          // recomputed S_hat
  hgm_fused_kernel<<<grid, block, smem_bytes, stream>>>(hs, ht, out_val, out_idx);
}